// TorchRoughSABR_FMM_17068200034436
// MI455X (gfx1250) — compile-verified
//
#include <hip/hip_runtime.h>
#include <hip/hip_bf16.h>
#include <math.h>

typedef __attribute__((ext_vector_type(2))) float v2f;
typedef __attribute__((ext_vector_type(8))) float v8f;

#define N_STEPS   512
#define N_FWD     40
#define DT_F      (1.0f / 512.0f)
#define ALPHA1    0.6                    // ALPHA + 1 = H - 0.5 + 1
#define GAMMA_06  1.4891922488128171     // Gamma(H + 0.5) = Gamma(0.6)

// ---- d_ws float layout (written by setup kernel, read by main kernel) ----
// WR is the REVERSED, ZERO-PADDED Toeplitz weight array:
//   WR[i] = w[511 - i]  (i < 512),   WR[512..527] = 0
// so  w[j] == WR[511 - j]  and any j < 0 lands in the zero pad -> free causal mask.
#define WS_WR    0      // 528
#define WS_CS    528    // 512: inclusive prefix sum of w^2
#define WS_E     1040   // 40 x float4: {nu*rho*L0, nu*rho*L1, nu*rho*L2, nu*sqrt(1-rho^2)}
#define WS_G     1200   // 40 x float4: {vs*L0, vs*L1, vs*L2, 0.5*nu^2*DT}
#define WS_M0DT  1360   // 40: mu0 * DT
#define WS_F0    1400   // 40
#define WS_TOTAL 1440

// ---- main-kernel dynamic LDS float layout ----
#define LDS_Z      0        // [512][16] : dz for 4 paths x 4 channels
#define LDS_FBM    8192     // [256][16] : fbm4 half-buffer (time-blocked)
#define LDS_P      12288    // WS_TOTAL params
#define LDS_FLOATS (LDS_P + WS_TOTAL)   // 13728 floats = 54912 B  (< 64 KB)

// ---------------------------------------------------------------------------
// Setup: closed-form Toeplitz kernel weights + fully-folded per-forward coeffs.
// ---------------------------------------------------------------------------
__global__ __launch_bounds__(512) void rsabr_setup(
    const float* __restrict__ F0, const float* __restrict__ alphas,
    const float* __restrict__ rhos, const float* __restrict__ nus,
    const float* __restrict__ tau, const float* __restrict__ L,
    const float* __restrict__ Lam, float* __restrict__ ws)
{
  __shared__ float sw[N_STEPS];
  __shared__ float somega[N_FWD];
  __shared__ float svs[N_FWD];
  const int j = threadIdx.x;

  // kernel[t,s] = w(t-s),  w(j) = ((j+1)^0.6 - j^0.6)/0.6 * DT^(-0.4)/Gamma(0.6)
  const double scale = pow(1.0 / 512.0, -0.4) / GAMMA_06;
  const double ew = (pow((double)(j + 1), ALPHA1) - pow((double)j, ALPHA1)) / ALPHA1;
  const float wj = (float)(ew * scale);
  sw[j] = wj;
  ws[WS_WR + (N_STEPS - 1 - j)] = wj;            // reversed
  if (j < 16) ws[WS_WR + N_STEPS + j] = 0.0f;    // zero pad (causal mask)

  if (j < N_FWD) {
    const float f0 = F0[j];
    const float vs = alphas[j] * sqrtf(fabsf(f0 + 0.02f));   // beta = 0.5
    svs[j] = vs;
    somega[j] = tau[j] * vs / (1.0f + tau[j] * f0);
  }
  __syncthreads();

  if (j == 0) {
    float run = 0.0f;
    for (int t = 0; t < N_STEPS; ++t) { run += sw[t] * sw[t]; ws[WS_CS + t] = run; }
  }
  if (j < N_FWD) {
    float acc = 0.0f;
    for (int m = 0; m < N_FWD; ++m) acc += Lam[j * N_FWD + m] * somega[m];
    const float rho = rhos[j], nu = nus[j], vs = svs[j];
    const float s1r = sqrtf(fmaxf(0.0f, 1.0f - rho * rho));
    const float nr  = nu * rho;
    // exponent coefficients: nu*fbm - vc*CS  ==  E . {f0,f1,f2,f3}  -  G.w*CS
    ws[WS_E + j*4 + 0] = nr * L[j*3 + 0];
    ws[WS_E + j*4 + 1] = nr * L[j*3 + 1];
    ws[WS_E + j*4 + 2] = nr * L[j*3 + 2];
    ws[WS_E + j*4 + 3] = nu * s1r;
    // wr * vol_scale coefficients (+ vc packed in .w)
    ws[WS_G + j*4 + 0] = vs * L[j*3 + 0];
    ws[WS_G + j*4 + 1] = vs * L[j*3 + 1];
    ws[WS_G + j*4 + 2] = vs * L[j*3 + 2];
    ws[WS_G + j*4 + 3] = 0.5f * nu * nu * DT_F;
    ws[WS_M0DT + j] = (-vs * acc) * DT_F;        // mu0 * DT
    ws[WS_F0   + j] = F0[j];
  }
}

// ---------------------------------------------------------------------------
// Main fused kernel: 4 paths / block, 256 threads (8 wave32).
//   stage1: dz -> LDS Z[512][16]
//   stage2: FBM = Toeplitz(K) @ Z via v_wmma_f32_16x16x4_f32; pointer-walk
//           k-loop: 2x ds_load_2addr + 1x v_wmma per tile, scalar loop control.
//   stage3: coefficient-folded SABR vol + serial cumsum, 160 lanes.
// ---------------------------------------------------------------------------
__global__ __launch_bounds__(256) void rsabr_main(
    const float* __restrict__ dz, const float* __restrict__ ws,
    float* __restrict__ out)
{
  extern __shared__ float lds[];
  float* Z   = lds + LDS_Z;
  float* FBM = lds + LDS_FBM;
  float* P   = lds + LDS_P;

  const int tid = threadIdx.x;
  const int p0  = blockIdx.x * 4;

  for (int i = tid; i < WS_TOTAL; i += 256) P[i] = ws[i];

  // Stage dz: Z[s][4*pl + ch]; one float4 per (path, step), coalesced.
  for (int it = 0; it < 8; ++it) {
    const int idx = it * 256 + tid;          // 0..2047
    const int pl  = idx >> 9;
    const int s   = idx & (N_STEPS - 1);
    const float4 v = reinterpret_cast<const float4*>(dz)[(size_t)(p0 + pl) * N_STEPS + s];
    float* zp = &Z[s * 16 + pl * 4];
    zp[0] = v.x; zp[1] = v.y; zp[2] = v.z; zp[3] = v.w;
  }
  __syncthreads();

  const int lane  = tid & 31;
  const int wv    = tid >> 5;                // wave id 0..7 (wave-uniform)
  const int mrow  = lane & 15;               // M row / N col within fragment
  const int khalf = (lane >> 4) << 1;        // K offset for lanes 16..31
  const int roff  = (lane >> 4) << 3;        // D row offset for lanes 16..31

  // Per-(path, forward) series state for the cumsum threads (5 full waves).
  float4 E = {0,0,0,0}, G = {0,0,0,0};
  float m0dt = 0.0f, Freg = 0.0f;
  float* op = nullptr;
  int pl = 0;
  if (tid < 4 * N_FWD) {
    pl = tid / N_FWD;
    const int n = tid - pl * N_FWD;
    E = *reinterpret_cast<const float4*>(&P[WS_E + n*4]);
    G = *reinterpret_cast<const float4*>(&P[WS_G + n*4]);
    m0dt = P[WS_M0DT + n];
    Freg = P[WS_F0 + n];
    op = out + (size_t)(p0 + pl) * 513 * N_FWD + n;
    op[0] = Freg;                            // output[:, 0, :] = F0
  }
  const float* WR = P + WS_WR;
  const float* CS = P + WS_CS;

  for (int half = 0; half < 2; ++half) {
    // ---- WMMA stage: FBM rows [256*half, 256*half+256).
    // Each wave does tiles {base+wv, base+15-wv}: triangular cost balances.
    #pragma unroll
    for (int sel = 0; sel < 2; ++sel) {
      // mt is wave-uniform; pin it into an SGPR so the k-loop is scalar.
      const int mt    = __builtin_amdgcn_readfirstlane(
                            half * 16 + (sel ? (15 - wv) : wv));
      const int mbase = mt << 4;
      const int row   = mbase + mrow;
      v8f c = {};

      // Pointer walk: A pair ascends through reversed/zero-padded WR in the
      // correct {a0,a1} order; B pair is {Z[kb][n], Z[kb+1][n]}.
      const float* wp = WR + (N_STEPS - 1 - row + khalf);
      const float* zp = Z + khalf * 16 + mrow;
      const int kktot = (mt << 2) + 4;       // all causal k-tiles incl. diagonal
      #pragma unroll 4
      for (int kk = 0; kk < kktot; ++kk) {
        const v2f A = { wp[0], wp[1] };
        const v2f B = { zp[0], zp[16] };
        wp += 4;
        zp += 64;
        c = __builtin_amdgcn_wmma_f32_16x16x4_f32(false, A, false, B,
                                                  (short)0, c, false, false);
      }

      const int lr = mbase - half * 256 + roff;   // local FBM row base
      #pragma unroll
      for (int r = 0; r < 8; ++r) FBM[(lr + r) * 16 + mrow] = c[r];
    }
    __syncthreads();

    // ---- elementwise SABR + cumsum over this half (160 active lanes).
    if (tid < 4 * N_FWD) {
      const float* fb = FBM + pl * 4;
      const float* zb = Z + pl * 4 + half * 256 * 16;
      float* o = op + (half * 256 + 1) * N_FWD;
      for (int t = 0; t < 256; ++t) {
        const float4 f4 = *reinterpret_cast<const float4*>(fb + t * 16);
        const float4 z4 = *reinterpret_cast<const float4*>(zb + t * 16);
        const float ex  = E.x*f4.x + E.y*f4.y + E.z*f4.z + E.w*f4.w
                          - G.w * CS[half * 256 + t];
        const float uv  = __expf(ex);
        const float wrs = G.x*z4.x + G.y*z4.y + G.z*z4.z;   // wr * vol_scale
        Freg += fmaf(m0dt * uv, uv, wrs * uv);
        o[t * N_FWD] = Freg;
      }
    }
    __syncthreads();   // FBM half-buffer reused by next half's WMMA stage
  }
}

// ---------------------------------------------------------------------------
extern "C" void kernel_launch(void* const* d_in, const int* in_sizes, int n_in,
                              void* d_out, int out_size, void* d_ws, size_t ws_size,
                              hipStream_t stream)
{
  const float* dz     = (const float*)d_in[0];   // (2048, 512, 4) f32
  const float* F0     = (const float*)d_in[1];   // (40,)
  const float* alphas = (const float*)d_in[2];   // (40,)
  const float* rhos   = (const float*)d_in[3];   // (40,)
  const float* nus    = (const float*)d_in[4];   // (40,)
  const float* tau    = (const float*)d_in[5];   // (40,)
  const float* L      = (const float*)d_in[6];   // (40, 3)
  const float* Lam    = (const float*)d_in[7];   // (40, 40)
  float* ws  = (float*)d_ws;
  float* out = (float*)d_out;                    // (2048, 513, 40) f32

  const int n_paths = in_sizes[0] / (N_STEPS * 4);

  rsabr_setup<<<1, N_STEPS, 0, stream>>>(F0, alphas, rhos, nus, tau, L, Lam, ws);
  rsabr_main<<<n_paths / 4, 256, LDS_FLOATS * sizeof(float), stream>>>(dz, ws, out);
}